// MPS_5317169512951
// MI455X (gfx1250) — compile-verified
//
#include <hip/hip_runtime.h>

// MPS batched amplitude evaluation for MI455X (gfx1250).
// One wave32 owns a 16-row batch tile; env (16x32 f32) lives in LDS.
// Per site: env @ [M0 | M1] via 32x V_WMMA_F32_16X16X4_F32 (8 K-steps x
// 4 N-tiles), then exact per-row v_cndmask select between the M0/M1 halves.
// Site matrix pair staged TRANSPOSED in double-buffered LDS so every WMMA
// operand fragment is one aligned ds_load_b64 (no register shuffles).

typedef __attribute__((ext_vector_type(2))) float v2f;
typedef __attribute__((ext_vector_type(8))) float v8f;

#define BATCH 32768
#define NSITES 64
#define BOND 32
#define ROWS_PER_WAVE 16
#define WAVES_PER_WG 4
#define ROWS_PER_WG (ROWS_PER_WAVE * WAVES_PER_WG)
#define RPAD 68   // sM stride (floats) over the 64 (p,d) rows: bank = 4*l16+2*hi
#define EPAD 34   // env row stride (floats): even -> aligned b64 A-fragments
#define CPAD 20   // cfg per-site stride (ints): 80B, 16B-aligned int4 reads

__global__ __launch_bounds__(WAVES_PER_WG * 32)
void mps_psi_kernel(const int*   __restrict__ cfg,     // [BATCH][NSITES]
                    const float* __restrict__ leftT,   // [2][32]
                    const float* __restrict__ bulkT,   // [62][32][2][32] (d,p,e)
                    const float* __restrict__ rightT,  // [32][2]
                    float*       __restrict__ out)     // [BATCH]
{
    // Transposed site matrices: sM[buf][e*RPAD + p*32 + d]
    __shared__ float sM[2][BOND * RPAD];
    __shared__ float sEnv[WAVES_PER_WG][ROWS_PER_WAVE * EPAD];
    __shared__ __align__(16) int sCfgAll[WAVES_PER_WG][NSITES][CPAD];

    const int tid   = threadIdx.x;
    const int wave  = tid >> 5;
    const int lane  = tid & 31;
    const int l16   = lane & 15;
    const int hi    = lane >> 4;                 // 0: lanes 0-15, 1: lanes 16-31
    const int rowBase = blockIdx.x * ROWS_PER_WG + wave * ROWS_PER_WAVE;

    float* env = &sEnv[wave][0];

    // ---- stage this wave's 16x64 config block (transposed: [site][row]) ----
    for (int i = lane; i < ROWS_PER_WAVE * NSITES; i += 32) {
        const int r = i >> 6;        // 0..15
        const int c = i & 63;        // site (coalesced on global side)
        sCfgAll[wave][c][r] = cfg[(rowBase + r) * NSITES + c];
    }
    // Per-wave data + in-order DS within a wave: no barrier needed here.

    // ---- init env[r][d] = left_tensor[cfg[r][0]][d] ----
    for (int i = lane; i < ROWS_PER_WAVE * BOND; i += 32) {
        const int r = i >> 5;
        const int d = i & 31;
        const int s0 = sCfgAll[wave][0][r];
        env[r * EPAD + d] = leftT[s0 * BOND + d];
    }

    for (int site = 1; site <= NSITES - 2; ++site) {
        const float* Msite = bulkT + (size_t)(site - 1) * (BOND * 2 * BOND);
        float* M = &sM[(site - 1) & 1][0];

        // Cooperative transposed load: M[e*RPAD + p*32 + d] = bulk[s-1][d][p][e]
        for (int i = tid; i < 2 * BOND * BOND; i += (WAVES_PER_WG * 32)) {
            const int e  = i & 31;       // coalesced on global side
            const int rp = i >> 5;       // p*32 + d
            M[e * RPAD + rp] = Msite[(rp & 31) * 64 + (rp >> 5) * 32 + e];
        }
        // Prefetch next site's matrix pair (global_prefetch_b8)
        if (site < NSITES - 2) {
            __builtin_prefetch(Msite + (BOND * 2 * BOND) + tid * 16, 0, 0);
        }
        __syncthreads();   // double-buffered: one barrier per site suffices

        // Spin bits for this site: rows 8*hi .. 8*hi+7 (broadcast b128 loads)
        int sarr[8];
        *(int4*)&sarr[0] = *(const int4*)&sCfgAll[wave][site][8 * hi];
        *(int4*)&sarr[4] = *(const int4*)&sCfgAll[wave][site][8 * hi + 4];

        // A fragments (32-bit A 16x4 layout): lanes 0-15 row=l16 K={4k,4k+1},
        // lanes 16-31 row=l16 K={4k+2,4k+3}. Aligned b64 loads.
        v2f a[8];
#pragma unroll
        for (int k = 0; k < 8; ++k) {
            a[k] = *(const v2f*)&env[l16 * EPAD + 4 * k + 2 * hi];
        }

        v8f acc[4] = {};   // N-tiles: cols [0..15|16..31] of M0, then of M1

#pragma unroll
        for (int k = 0; k < 8; ++k) {
            const int r0b = 4 * k + 2 * hi;     // K rows {r0b, r0b+1} per lane
#pragma unroll
            for (int n = 0; n < 4; ++n) {
                // B element j = M[d = r0b+j][p = n>>1][e = 16*(n&1)+l16]
                // transposed LDS -> the j=0,1 pair is contiguous: one b64.
                const int colg = (n & 1) * 16 + l16;
                const int prow = (n >> 1) * 32;
                const v2f b = *(const v2f*)&M[colg * RPAD + prow + r0b];
                acc[n] = __builtin_amdgcn_wmma_f32_16x16x4_f32(
                    false, a[k], false, b, (short)0, acc[n], false, false);
            }
        }

        // ---- exact per-row select (v_cndmask) between M0/M1 results.
        // D layout: acc[n][v] @ lane l == row v+8*hi, col (n&1)*16 + l16.
#pragma unroll
        for (int v = 0; v < 8; ++v) {
            const int row = v + 8 * hi;
            const bool sp = (sarr[v] != 0);
            env[row * EPAD + l16]      = sp ? acc[2][v] : acc[0][v];
            env[row * EPAD + 16 + l16] = sp ? acc[3][v] : acc[1][v];
        }
    }

    // ---- final contraction with the selected right_tensor column ----
    if (lane < ROWS_PER_WAVE) {
        const int r = rowBase + lane;
        const int p = sCfgAll[wave][NSITES - 1][lane];
        float acc = 0.0f;
#pragma unroll 8
        for (int d = 0; d < BOND; ++d) {
            acc = fmaf(env[lane * EPAD + d], rightT[d * 2 + p], acc);
        }
        out[r] = acc;
    }
}

extern "C" void kernel_launch(void* const* d_in, const int* in_sizes, int n_in,
                              void* d_out, int out_size, void* d_ws, size_t ws_size,
                              hipStream_t stream) {
    const int*   cfg    = (const int*)  d_in[0];
    const float* leftT  = (const float*)d_in[1];
    const float* bulkT  = (const float*)d_in[2];
    const float* rightT = (const float*)d_in[3];
    float*       out    = (float*)d_out;
    (void)in_sizes; (void)n_in; (void)out_size; (void)d_ws; (void)ws_size;

    dim3 grid(BATCH / ROWS_PER_WG);        // 512 workgroups
    dim3 block(WAVES_PER_WG * 32);         // 128 threads = 4 waves (wave32)
    hipLaunchKernelGGL(mps_psi_kernel, grid, block, 0, stream,
                       cfg, leftT, bulkT, rightT, out);
}